// Similarity_16303695856108
// MI455X (gfx1250) — compile-verified
//
#include <hip/hip_runtime.h>

typedef float v2f __attribute__((ext_vector_type(2)));
typedef float v4f __attribute__((ext_vector_type(4)));
typedef float v8f __attribute__((ext_vector_type(8)));

#define B_   64
#define T_   1024
#define J_   256
#define D2_  256

#define TILE    64               // T-tile and J-tile per block
#define KCHUNK  64               // K staged per LDS pass
#define LSTRIDE (KCHUNK + 4)     // pad to avoid LDS bank conflicts on column reads
#define BUFELEM (TILE * LSTRIDE)

// gfx1250 flat-shared aperture: addr[63:32] selects LDS, addr[31:0] is the LDS
// byte offset -> truncation of a generic __shared__ pointer gives the DS address.
__device__ __forceinline__ unsigned lds_off(const void* p) {
    return (unsigned)(unsigned long long)p;
}

// One 16-byte async global->LDS copy (per-lane), tracked by ASYNCcnt.
__device__ __forceinline__ void async_copy_b128(const float* gsrc, float* ldst) {
    unsigned l = lds_off(ldst);
    asm volatile("global_load_async_to_lds_b128 %0, %1, off"
                 :: "v"(l), "v"(gsrc) : "memory");
}

__device__ __forceinline__ void wait_async0() {
    asm volatile("s_wait_asynccnt 0x0" ::: "memory");
}

__global__ __launch_bounds__(256)
void sim_s_kernel(const float* __restrict__ H, const float* __restrict__ U,
                  const float* __restrict__ w, const float* __restrict__ bptr,
                  float* __restrict__ S)
{
    __shared__ float HsB[2][BUFELEM];   // H tile, pre-scaled by w_hu (ping-pong)
    __shared__ float UsB[2][BUFELEM];   // U tile, raw (async-filled, ping-pong)
    __shared__ float sh[TILE];          // s_h per t-row
    __shared__ float su[TILE];          // s_u per j-row
    __shared__ float wbuf[3 * D2_];     // w_h | w_u | w_hu

    const int tid  = threadIdx.x;
    const int lane = tid & 31;
    const int wave = tid >> 5;
    const int wt   = wave >> 1;         // wave row-tile  (0..3) -> 16 rows
    const int wj   = wave & 1;          // wave col-group (0..1) -> 32 cols

    const int bidx = blockIdx.z;
    const int t0   = blockIdx.y * TILE;
    const int j0   = blockIdx.x * TILE;

    for (int i = tid; i < 3 * D2_; i += 256) wbuf[i] = w[i];
    if (tid < TILE) { sh[tid] = 0.0f; su[tid] = 0.0f; }
    __syncthreads();

    const float bias = bptr[0];

    v8f acc0 = {0.f,0.f,0.f,0.f,0.f,0.f,0.f,0.f};
    v8f acc1 = {0.f,0.f,0.f,0.f,0.f,0.f,0.f,0.f};

    // Loader mapping: 4 threads per row, 16 consecutive floats (4x b128) each.
    const int lrow = tid >> 2;          // 0..63
    const int lcol = (tid & 3) * 16;    // 0,16,32,48

    const float* Hrow = H + ((size_t)bidx * T_ + (size_t)(t0 + lrow)) * D2_;
    const float* Urow = U + ((size_t)bidx * J_ + (size_t)(j0 + lrow)) * D2_;

    // WMMA fragment addressing (32-bit A 16x4 layout):
    //  lanes 0-15 : row = lane,    K = {k+0,k+1};  lanes 16-31: row = lane-16, K = {k+2,k+3}
    const int fr = lane & 15;
    const int ko = (lane >> 4) * 2;

    // ---- prologue: stage chunk 0 into buffer 0 ----
    {
        #pragma unroll
        for (int v = 0; v < 4; ++v)
            async_copy_b128(Urow + lcol + v * 4,
                            &UsB[0][lrow * LSTRIDE + lcol + v * 4]);
        float hpart = 0.0f;
        #pragma unroll
        for (int v = 0; v < 4; ++v) {
            v4f hv = *(const v4f*)(Hrow + lcol + v * 4);
            #pragma unroll
            for (int e = 0; e < 4; ++e) {
                const int c = lcol + v * 4 + e;
                hpart += hv[e] * wbuf[c];                       // w_h fold
                HsB[0][lrow * LSTRIDE + c] = hv[e] * wbuf[2 * D2_ + c];  // w_hu fold
            }
        }
        atomicAdd(&sh[lrow], hpart);
        wait_async0();
        __syncthreads();
    }

    // ---- main loop: compute buf[cur] while staging buf[nxt] ----
    #pragma unroll
    for (int kc = 0; kc < D2_; kc += KCHUNK) {
        const int  cur      = (kc / KCHUNK) & 1;
        const int  nxt      = cur ^ 1;
        const int  kn       = kc + KCHUNK;
        const bool has_next = (kn < D2_);

        // Issue next-chunk staging early (async U; H global loads into regs).
        v4f hvr[4];
        if (has_next) {
            #pragma unroll
            for (int v = 0; v < 4; ++v)
                async_copy_b128(Urow + kn + lcol + v * 4,
                                &UsB[nxt][lrow * LSTRIDE + lcol + v * 4]);
            #pragma unroll
            for (int v = 0; v < 4; ++v)
                hvr[v] = *(const v4f*)(Hrow + kn + lcol + v * 4);
        }

        // s_u rank-1 partials from the staged raw U tile.
        {
            float upart = 0.0f;
            const float* ub = &UsB[cur][lrow * LSTRIDE + lcol];
            #pragma unroll
            for (int e = 0; e < 16; ++e)
                upart += ub[e] * wbuf[D2_ + kc + lcol + e];
            atomicAdd(&su[lrow], upart);
        }

        // WMMA sweep over the chunk: D = A x B + C, f32 16x16x4.
        const float* Abase  = &HsB[cur][(wt * 16      + fr) * LSTRIDE + ko];
        const float* B0base = &UsB[cur][(wj * 32      + fr) * LSTRIDE + ko];
        const float* B1base = &UsB[cur][(wj * 32 + 16 + fr) * LSTRIDE + ko];
        #pragma unroll
        for (int k = 0; k < KCHUNK; k += 4) {
            v2f a  = *(const v2f*)(Abase  + k);
            v2f b0 = *(const v2f*)(B0base + k);
            v2f b1 = *(const v2f*)(B1base + k);
            acc0 = __builtin_amdgcn_wmma_f32_16x16x4_f32(
                       false, a, false, b0, (short)0, acc0, false, false);
            acc1 = __builtin_amdgcn_wmma_f32_16x16x4_f32(
                       false, a, false, b1, (short)0, acc1, false, false);
        }

        // Fold w_hu into next H chunk, accumulate s_h, store into the other buffer.
        if (has_next) {
            float hpart = 0.0f;
            #pragma unroll
            for (int v = 0; v < 4; ++v) {
                #pragma unroll
                for (int e = 0; e < 4; ++e) {
                    const int c = lcol + v * 4 + e;
                    hpart += hvr[v][e] * wbuf[kn + c - lcol + lcol]; // w_h[kn + c]
                    HsB[nxt][lrow * LSTRIDE + c] = hvr[v][e] * wbuf[2 * D2_ + kn + c];
                }
            }
            atomicAdd(&sh[lrow], hpart);
            wait_async0();
        }
        __syncthreads();
    }

    // ---- epilogue: S = acc + s_h[t] + s_u[j] + bias, streamed non-temporal ----
    // NT stores keep the 64 MB output stream out of L2 so the reused 80 MB
    // H+U working set stays resident (H re-read by 4 j-blocks, U by 16 t-blocks).
    const int col  = lane & 15;
    const int rsel = (lane >> 4) * 8;
    const float su0 = su[wj * 32      + col];
    const float su1 = su[wj * 32 + 16 + col];
    #pragma unroll
    for (int v = 0; v < 8; ++v) {
        const int r = wt * 16 + rsel + v;
        const float rowterm = sh[r] + bias;
        const size_t o = ((size_t)bidx * T_ + (size_t)(t0 + r)) * J_
                       + (size_t)(j0 + wj * 32 + col);
        __builtin_nontemporal_store(acc0[v] + rowterm + su0, &S[o]);
        __builtin_nontemporal_store(acc1[v] + rowterm + su1, &S[o + 16]);
    }
}

extern "C" void kernel_launch(void* const* d_in, const int* in_sizes, int n_in,
                              void* d_out, int out_size, void* d_ws, size_t ws_size,
                              hipStream_t stream) {
    (void)in_sizes; (void)n_in; (void)out_size; (void)d_ws; (void)ws_size;
    const float* H    = (const float*)d_in[0];   // (64, 1024, 256) f32
    const float* U    = (const float*)d_in[1];   // (64,  256, 256) f32
    const float* w    = (const float*)d_in[2];   // (768,) f32
    const float* bptr = (const float*)d_in[3];   // scalar f32
    float*       S    = (float*)d_out;           // (64, 1024, 256) f32

    dim3 grid(J_ / TILE, T_ / TILE, B_);         // (4, 16, 64)
    dim3 block(256);
    sim_s_kernel<<<grid, block, 0, stream>>>(H, U, w, bptr, S);
}